// Loss_NormalCrossVert_2516850835619
// MI455X (gfx1250) — compile-verified
//
#include <hip/hip_runtime.h>

// Loss_NormalCrossVert on MI455X (gfx1250, wave32).
//
// loss = (1/R) * sum_e ( dot(normal[e/16], v[e0]-v[e1])^2 / |v[e0]-v[e1]|^2 )
//
// Pass 1: grid-stride gather/accumulate, wave32 shuffle reduce, one fp32
//         partial per block into d_ws.
// Pass 2: single wave reduces partials; the 32->1 lane reduction is done with
//         V_WMMA_F32_16X16X4_F32 (A = lane sums, B = ones) — fp32 matrix path.

#define NUM_V   500000
#define DEG     16
#define R_EDGES (NUM_V * DEG)
#define BLOCK   256

typedef float v2f __attribute__((ext_vector_type(2)));
typedef float v8f __attribute__((ext_vector_type(8)));

__global__ __launch_bounds__(BLOCK) void edge_loss_partial(
    const float* __restrict__ nrm,      // [NUM_V*3]
    const float* __restrict__ vert,     // [NUM_V*3]
    const int*   __restrict__ e0,       // [R]
    const int*   __restrict__ e1,       // [R]
    const int*   __restrict__ vids,     // [R] == e >> 4
    float*       __restrict__ partial,  // [gridDim.x]
    int totalThreads)
{
    const int tid = blockIdx.x * BLOCK + threadIdx.x;

    float acc = 0.0f;
    // Grid-stride: consecutive lanes hit consecutive edges -> coalesced index
    // loads; the vertex gathers are random but the 6 MB table lives in L2.
    for (int e = tid; e < R_EDGES; e += totalThreads) {
        const int i0 = e0[e] * 3;
        const int i1 = e1[e] * 3;
        const int iv = vids[e] * 3;

        const float dx = vert[i0 + 0] - vert[i1 + 0];
        const float dy = vert[i0 + 1] - vert[i1 + 1];
        const float dz = vert[i0 + 2] - vert[i1 + 2];

        const float len2 = dx * dx + dy * dy + dz * dz;
        const float dot  = nrm[iv + 0] * dx + nrm[iv + 1] * dy + nrm[iv + 2] * dz;

        // (dot/sqrt(len2))^2 == dot*dot/len2 : kill the sqrt entirely.
        acc += (dot * dot) / len2;
    }

    // wave32 butterfly reduction
    #pragma unroll
    for (int off = 16; off > 0; off >>= 1)
        acc += __shfl_xor(acc, off, 32);

    __shared__ float smem[BLOCK / 32];
    if ((threadIdx.x & 31) == 0)
        smem[threadIdx.x >> 5] = acc;
    __syncthreads();

    if (threadIdx.x == 0) {
        float b = 0.0f;
        #pragma unroll
        for (int w = 0; w < BLOCK / 32; ++w)
            b += smem[w];
        partial[blockIdx.x] = b;
    }
}

__global__ __launch_bounds__(32) void reduce_final(
    const float* __restrict__ partial, int n, float* __restrict__ out)
{
    const int lane = threadIdx.x;

    float s = 0.0f;
    for (int i = lane; i < n; i += 32)
        s += partial[i];

    // 32 lane sums -> 1 via V_WMMA_F32_16X16X4_F32.
    // A (16x4 f32, 2 VGPRs): vgpr0 = s  => A[m][0]=s[m], A[m][2]=s[m+16],
    // vgpr1 = 0. B = ones(4x16). D[m][n] = s[m] + s[m+16] for every n.
    v2f a; a[0] = s;    a[1] = 0.0f;
    v2f b; b[0] = 1.0f; b[1] = 1.0f;
    v8f c = {};
    v8f d = __builtin_amdgcn_wmma_f32_16x16x4_f32(
        /*neg_a=*/false, a, /*neg_b=*/false, b,
        /*c_mod=*/(short)0, c, /*reuse_a=*/false, /*reuse_b=*/false);

    // lane L (L<16) holds D[0..7][L]; lane L+16 holds D[8..15][L].
    float t = d[0] + d[1] + d[2] + d[3] + d[4] + d[5] + d[6] + d[7];
    t += __shfl_xor(t, 16, 32);   // rows 0..7 + rows 8..15 = all 32 lane sums

    if (lane == 0)
        out[0] = t / (float)R_EDGES;
}

extern "C" void kernel_launch(void* const* d_in, const int* in_sizes, int n_in,
                              void* d_out, int out_size, void* d_ws, size_t ws_size,
                              hipStream_t stream)
{
    const float* nrm  = (const float*)d_in[0];  // normalArray [NUM_V,3] f32
    const float* vert = (const float*)d_in[1];  // vertArray   [NUM_V,3] f32
    const int*   e0   = (const int*)d_in[2];    // edge_0 [R] i32
    const int*   e1   = (const int*)d_in[3];    // edge_1 [R] i32
    const int*   vids = (const int*)d_in[4];    // vert_ids [R] i32
    float*       out  = (float*)d_out;
    float*       part = (float*)d_ws;

    // ~8 edges per thread by default; clamp block count to available workspace.
    int nblk = (R_EDGES / 8 + BLOCK - 1) / BLOCK;      // 3907
    const int maxblk = (int)(ws_size / sizeof(float));
    if (nblk > maxblk) nblk = maxblk > 0 ? maxblk : 1;

    const int totalThreads = nblk * BLOCK;

    edge_loss_partial<<<nblk, BLOCK, 0, stream>>>(nrm, vert, e0, e1, vids,
                                                  part, totalThreads);
    reduce_final<<<1, 32, 0, stream>>>(part, nblk, out);
}